// DropStripes_79267916415506
// MI455X (gfx1250) — compile-verified
//
#include <hip/hip_runtime.h>

// DropStripes: out[b,0,t,f] = x[b,0,t,f] * keep[b,t]
//   keep[b,t] = 0 iff t in [bgn[b,s], bgn[b,s]+dist[b,s]) for s in {0,1}
// B=64, C=1, T=4096, F=256, float32.
// Pure streaming op: 512 MiB traffic -> ~23 us floor at 23.3 TB/s.
// Use the CDNA5 async global->LDS path (ASYNCcnt) + NT 128-bit stores.

#define B_DIM 64
#define T_DIM 4096
#define F_DIM 256

#define ROWS_PER_BLOCK 8                      // 8 rows * 256 f32 = 8 KiB LDS
#define BLOCKS_PER_B  (T_DIM / ROWS_PER_BLOCK) // 512
#define V4_PER_BLOCK  (ROWS_PER_BLOCK * F_DIM / 4) // 512 float4 tiles

typedef __attribute__((ext_vector_type(4))) float v4f;
typedef __attribute__((address_space(3))) v4f lds_v4f;

__global__ __launch_bounds__(256) void drop_stripes_kernel(
    const float* __restrict__ x,
    const int*   __restrict__ bgn,
    const int*   __restrict__ dist,
    float*       __restrict__ out)
{
    __shared__ v4f tile[V4_PER_BLOCK];

    const int tid = threadIdx.x;
    const int bid = blockIdx.x;

    // b, t0 uniform per block -> stripe params become scalar loads
    const int b  = bid / BLOCKS_PER_B;
    const int t0 = (bid % BLOCKS_PER_B) * ROWS_PER_BLOCK;

    const int b0 = bgn[b * 2 + 0];
    const int b1 = bgn[b * 2 + 1];
    const int e0 = b0 + dist[b * 2 + 0];
    const int e1 = b1 + dist[b * 2 + 1];

    const size_t base = ((size_t)b * T_DIM + (size_t)t0) * F_DIM; // in floats

    // ---- Stage 1: async-copy 8 rows (8 KiB) global -> LDS, b128 per lane ----
    // Each lane issues 2 async loads; it will consume exactly the LDS bytes it
    // requested, so only the wave-private ASYNCcnt wait is needed (no barrier).
#pragma unroll
    for (int k = 0; k < 2; ++k) {
        const int idx = tid + (k << 8);                 // 0..511
        // 32-bit LDS byte offset of the destination (addrspace(3) ptr value)
        unsigned loff = (unsigned)(size_t)(lds_v4f*)&tile[idx];
        const float* gp = x + base + ((size_t)idx << 2);
        asm volatile("global_load_async_to_lds_b128 %0, %1, off"
                     :: "v"(loff), "v"(gp)
                     : "memory");
    }
    asm volatile("s_wait_asynccnt 0x0" ::: "memory");

    // ---- Stage 2: LDS -> mask -> non-temporal b128 store ----
#pragma unroll
    for (int k = 0; k < 2; ++k) {
        const int idx = tid + (k << 8);
        const int t   = t0 + (idx >> 6);                // 64 float4 per row
        const bool drop = ((t >= b0) & (t < e0)) | ((t >= b1) & (t < e1));
        const float keep = drop ? 0.0f : 1.0f;

        v4f v = tile[idx];
        v = v * keep;

        // Streamed once, dataset >> L2: non-temporal store (th:NT)
        __builtin_nontemporal_store(v, (v4f*)(out + base + ((size_t)idx << 2)));
    }
}

extern "C" void kernel_launch(void* const* d_in, const int* in_sizes, int n_in,
                              void* d_out, int out_size, void* d_ws, size_t ws_size,
                              hipStream_t stream)
{
    const float* x    = (const float*)d_in[0]; // (B, C, T, F) f32
    const int*   bgn  = (const int*)d_in[1];   // (B, 2) i32
    const int*   dist = (const int*)d_in[2];   // (B, 2) i32
    float*       out  = (float*)d_out;         // (B, C, T, F) f32

    dim3 grid(B_DIM * BLOCKS_PER_B);           // 32768 blocks
    dim3 block(256);                           // 8 waves (wave32)
    hipLaunchKernelGGL(drop_stripes_kernel, grid, block, 0, stream,
                       x, bgn, dist, out);
}